// DeepseekV3TopkRouter_4501125726820
// MI455X (gfx1250) — compile-verified
//
#include <hip/hip_runtime.h>
#include <hip/hip_bf16.h>

// Problem dims (fixed by reference): B=4, S=4096 -> M = 16384 rows, K = H = 4096, E = 64, top_k = 8
#define M_TOT  16384
#define K_DIM  4096
#define E_DIM  64
#define TOPK   8
#define KC     64            // K-chunk staged in LDS
#define LDSTR  (KC + 4)      // padded row stride (floats): 68 -> bank-conflict-free fragments

typedef float v2f __attribute__((ext_vector_type(2)));
typedef float v8f __attribute__((ext_vector_type(8)));

// ---------------------------------------------------------------------------
// Async-prefetch one K-chunk of the weight matrix into an LDS buffer.
// 256 threads; thread t copies 16 floats of expert row e = t>>2 via four
// global_load_async_to_lds_b128 (ASYNCcnt-tracked, CDNA5 async path).
// ---------------------------------------------------------------------------
__device__ __forceinline__ void prefetch_w_chunk(const float* __restrict__ W,
                                                 float (*buf)[LDSTR],
                                                 int kc, int tid) {
    const int e   = tid >> 2;          // expert row 0..63
    const int seg = (tid & 3) * 16;    // 16-float segment within chunk
    const float* g = W + (size_t)e * K_DIM + kc * KC + seg;
    uint32_t l = (uint32_t)(uintptr_t)(&buf[e][seg]);
    #pragma unroll
    for (int p = 0; p < 4; ++p) {
        asm volatile("global_load_async_to_lds_b128 %0, %1, off"
                     :: "v"(l + p * 16), "v"(g + p * 4)
                     : "memory");
    }
}

// ---------------------------------------------------------------------------
// GEMM: logits = hidden @ weight.T   using V_WMMA_F32_16X16X4_F32 (wave32)
// 128 blocks x 256 threads (8 waves). Wave = 16 rows x 64 experts.
// Weight double-buffered in LDS via async loads; A streamed from HBM.
// ---------------------------------------------------------------------------
__global__ __launch_bounds__(256)
void router_gemm_kernel(const float* __restrict__ A,   // [M_TOT, K_DIM]
                        const float* __restrict__ W,   // [E_DIM, K_DIM]
                        float* __restrict__ out)       // [M_TOT, E_DIM]
{
    __shared__ float sw[2][E_DIM][LDSTR];   // 2 * 64 * 68 * 4 = 34.8 KB

    const int tid  = threadIdx.x;
    const int lane = tid & 31;
    const int wave = tid >> 5;
    const int row_base = (blockIdx.x * 8 + wave) * 16;

    const int half = lane >> 4;   // 0: lanes 0-15, 1: lanes 16-31
    const int m    = lane & 15;

    // A fragment (16x4 fp32): lane (half,m) supplies row m, cols {k+2*half, k+2*half+1}
    const float* arow = A + (size_t)(row_base + m) * K_DIM + 2 * half;

    v8f acc0 = {}, acc1 = {}, acc2 = {}, acc3 = {};

    prefetch_w_chunk(W, sw[0], 0, tid);

    for (int c = 0; c < K_DIM / KC; ++c) {
        // drain this wave's async fills, then make all waves' fills visible
        asm volatile("s_wait_asynccnt 0x0" ::: "memory");
        __syncthreads();

        if (c + 1 < K_DIM / KC)
            prefetch_w_chunk(W, sw[(c + 1) & 1], c + 1, tid);

        const float (*bw)[LDSTR] = sw[c & 1];
        const float* ac = arow + c * KC;

        #pragma unroll 4
        for (int k = 0; k < KC; k += 4) {
            v2f a  = *(const v2f*)(ac + k);
            // B fragment (4x16): lane (half,m) -> col n = nt*16+m, rows {k+2h, k+2h+1}
            v2f b0 = *(const v2f*)(&bw[m +  0][k + 2 * half]);
            v2f b1 = *(const v2f*)(&bw[m + 16][k + 2 * half]);
            v2f b2 = *(const v2f*)(&bw[m + 32][k + 2 * half]);
            v2f b3 = *(const v2f*)(&bw[m + 48][k + 2 * half]);
            acc0 = __builtin_amdgcn_wmma_f32_16x16x4_f32(false, a, false, b0, (short)0, acc0, false, false);
            acc1 = __builtin_amdgcn_wmma_f32_16x16x4_f32(false, a, false, b1, (short)0, acc1, false, false);
            acc2 = __builtin_amdgcn_wmma_f32_16x16x4_f32(false, a, false, b2, (short)0, acc2, false, false);
            acc3 = __builtin_amdgcn_wmma_f32_16x16x4_f32(false, a, false, b3, (short)0, acc3, false, false);
        }

        __syncthreads();   // all waves done reading this buffer before its refill
    }

    // C/D layout: VGPR r holds M = r (lanes 0-15) / M = r+8 (lanes 16-31), N = lane&15
    float* orow = out + (size_t)row_base * E_DIM;
    #pragma unroll
    for (int r = 0; r < 8; ++r) {
        const int mo = r + 8 * half;
        float* p = orow + (size_t)mo * E_DIM + m;
        p[0]  = acc0[r];
        p[16] = acc1[r];
        p[32] = acc2[r];
        p[48] = acc3[r];
    }
}

// ---------------------------------------------------------------------------
// Top-8 + softmax over E=64 logits per row. One wave32 per row.
// ---------------------------------------------------------------------------
__global__ __launch_bounds__(256)
void topk_softmax_kernel(const float* __restrict__ logits,  // [M_TOT, E_DIM]
                         int* __restrict__ out_idx,         // [M_TOT, TOPK]
                         float* __restrict__ out_w)         // [M_TOT, TOPK]
{
    const int lane = threadIdx.x & 31;
    const int wave = threadIdx.x >> 5;
    const int row  = blockIdx.x * 8 + wave;
    if (row >= M_TOT) return;

    const float* lrow = logits + (size_t)row * E_DIM;
    float v0 = lrow[lane];
    float v1 = lrow[lane + 32];

    float topv[TOPK];
    int   topi[TOPK];

    #pragma unroll
    for (int it = 0; it < TOPK; ++it) {
        float bv; int bi;
        if (v0 >= v1) { bv = v0; bi = lane; }
        else          { bv = v1; bi = lane + 32; }
        #pragma unroll
        for (int off = 16; off > 0; off >>= 1) {
            float ov = __shfl_xor(bv, off, 32);
            int   oi = __shfl_xor(bi, off, 32);
            if (ov > bv || (ov == bv && oi < bi)) { bv = ov; bi = oi; }
        }
        topv[it] = bv;
        topi[it] = bi;
        if (bi == lane)      v0 = -__builtin_huge_valf();
        if (bi == lane + 32) v1 = -__builtin_huge_valf();
    }

    const float mx = topv[0];
    float e[TOPK];
    float s = 0.0f;
    #pragma unroll
    for (int i = 0; i < TOPK; ++i) { e[i] = __expf(topv[i] - mx); s += e[i]; }
    const float inv = 1.0f / s;

    if (lane == 0) {
        #pragma unroll
        for (int i = 0; i < TOPK; ++i) {
            out_idx[(size_t)row * TOPK + i] = topi[i];
            out_w [(size_t)row * TOPK + i] = e[i] * inv;
        }
    }
}

// ---------------------------------------------------------------------------
// d_out layout: [logits: M_TOT*E_DIM f32][indices: M_TOT*TOPK i32][weights: M_TOT*TOPK f32]
// ---------------------------------------------------------------------------
extern "C" void kernel_launch(void* const* d_in, const int* in_sizes, int n_in,
                              void* d_out, int out_size, void* d_ws, size_t ws_size,
                              hipStream_t stream) {
    const float* hidden = (const float*)d_in[0];  // (4, 4096, 4096) f32
    const float* weight = (const float*)d_in[1];  // (64, 4096) f32
    // d_in[2] = top_k scalar (fixed at 8)

    float* logits = (float*)d_out;
    int*   idx    = (int*)((float*)d_out + (size_t)M_TOT * E_DIM);
    float* wts    = (float*)d_out + (size_t)M_TOT * E_DIM + (size_t)M_TOT * TOPK;

    // GEMM: 16384 rows / (8 waves * 16 rows) = 128 blocks of 256 threads
    router_gemm_kernel<<<128, 256, 0, stream>>>(hidden, weight, logits);

    // Top-k: 8 rows (waves) per block
    topk_softmax_kernel<<<M_TOT / 8, 256, 0, stream>>>(logits, idx, wts);
}